// CTCLossLayer_25786983645673
// MI455X (gfx1250) — compile-verified
//
#include <hip/hip_runtime.h>
#include <stdint.h>

// Problem constants (match reference)
#define Bb    128
#define Tt    1024
#define Vv    256
#define Ll    128
#define Ss    257          // 2L+1 extended states
#define BLANKC 255
#define NEGF  (-1.0e30f)
#define NS    9            // states per lane: 32*9 = 288 >= 257
#define CH    8            // rows per TDM chunk (8 KB tile)
#define NCH   (Tt / CH)    // 128 chunks

typedef float        v2f __attribute__((ext_vector_type(2)));
typedef float        v8f __attribute__((ext_vector_type(8)));
typedef unsigned int v4u __attribute__((ext_vector_type(4)));
typedef int          v4i __attribute__((ext_vector_type(4)));
typedef int          v8i __attribute__((ext_vector_type(8)));

#if __has_builtin(__builtin_amdgcn_s_wait_tensorcnt)
#define WAIT_TENSORCNT(n) __builtin_amdgcn_s_wait_tensorcnt(n)
#else
#define WAIT_TENSORCNT(n) asm volatile("s_wait_tensorcnt %0" ::"i"(n) : "memory")
#endif

// ---------------------------------------------------------------------------
// Kernel 1: per-row logsumexp of y_pred rows (B*T rows of V=256 floats).
// One wave per row; streaming b128 loads; WMMA f32 16x16x4 folds the 32
// per-lane exp-partials to 16 in one instruction (A = ones, B vgpr0 = p,
// B vgpr1 = 0  =>  D[m][n] = p[n] + p[n+16] in element 0, all lanes).
// ---------------------------------------------------------------------------
__global__ __launch_bounds__(256) void ctc_row_lse(const float* __restrict__ yp,
                                                   float* __restrict__ rowOff) {
  const int wave = threadIdx.x >> 5;
  const int lane = threadIdx.x & 31;
  const int row  = blockIdx.x * 8 + wave;            // < B*T exactly

  const float4* rp = (const float4*)(yp + (size_t)row * Vv);
  float4 x0 = rp[lane];
  float4 x1 = rp[lane + 32];

  float m = fmaxf(fmaxf(fmaxf(x0.x, x0.y), fmaxf(x0.z, x0.w)),
                  fmaxf(fmaxf(x1.x, x1.y), fmaxf(x1.z, x1.w)));
  #pragma unroll
  for (int d = 16; d >= 1; d >>= 1) m = fmaxf(m, __shfl_xor(m, d, 32));

  float p = __expf(x0.x - m) + __expf(x0.y - m) + __expf(x0.z - m) + __expf(x0.w - m)
          + __expf(x1.x - m) + __expf(x1.y - m) + __expf(x1.z - m) + __expf(x1.w - m);

  // WMMA fold: 32 partials -> 16 (p[l&15] + p[(l&15)+16] in element 0, all lanes)
  v2f aOnes; aOnes[0] = 1.0f; aOnes[1] = 1.0f;
  v2f bP;    bP[0]    = p;    bP[1]    = 0.0f;
  v8f cz = {0.f, 0.f, 0.f, 0.f, 0.f, 0.f, 0.f, 0.f};
  v8f d = __builtin_amdgcn_wmma_f32_16x16x4_f32(false, aOnes, false, bP,
                                                (short)0, cz, false, false);
  float s = d[0];
  #pragma unroll
  for (int dd = 8; dd >= 1; dd >>= 1) s += __shfl_xor(s, dd, 32);

  if (lane == 0) rowOff[row] = m + __logf(s);
}

// ---------------------------------------------------------------------------
// TDM helper: DMA a CH x 256 f32 tile (global, row-major, stride 256) into LDS.
// Descriptor packed per CDNA5 ISA 8.3/8.4 (2D tile, groups 2/3/extra zero).
// ---------------------------------------------------------------------------
__device__ __forceinline__ unsigned lds_off(const void* p) {
  return (unsigned)(uintptr_t)p;     // generic LDS addr: low 32 bits = offset
}

__device__ __forceinline__ void tdm_load_rows(const float* gaddr, unsigned ldsByteOff) {
  uint64_t ga = (uint64_t)(uintptr_t)gaddr;
  v4u g0;
  g0[0] = 1u;                                            // count=1, no gather
  g0[1] = ldsByteOff;                                    // lds_addr
  g0[2] = (unsigned)(ga & 0xFFFFFFFFu);                  // global_addr[31:0]
  g0[3] = (unsigned)((ga >> 32) & 0x01FFFFFFu) | (2u << 30); // addr[56:32] | type=2
  v8i g1;
  g1[0] = (int)(2u << 16);           // workgroup_mask=0, data_size=2 (4B)
  g1[1] = (int)(256u << 16);         // tensor_dim0[15:0] = 256  (bits 63:48)
  g1[2] = (int)((unsigned)CH << 16); // tensor_dim0 hi=0; tensor_dim1[15:0]=CH (bits 95:80)
  g1[3] = (int)(256u << 16);         // tensor_dim1 hi=0; tile_dim0=256 (bits 127:112)
  g1[4] = CH;                        // tile_dim1=CH; tile_dim2=0
  g1[5] = 256;                       // tensor_dim0_stride low 32 = 256
  g1[6] = 0;                         // stride hi, tensor_dim1_stride low
  g1[7] = 0;
  v4i z4; z4[0] = 0; z4[1] = 0; z4[2] = 0; z4[3] = 0;
  v8i z8; z8[0] = 0; z8[1] = 0; z8[2] = 0; z8[3] = 0;
          z8[4] = 0; z8[5] = 0; z8[6] = 0; z8[7] = 0;
  __builtin_amdgcn_tensor_load_to_lds(g0, g1, z4, z4, z8, 0);
}

// ---------------------------------------------------------------------------
// Kernel 2: CTC alpha recursion. ONE wave per batch element:
//  - 9 states/lane in VGPRs (no barriers; shifts via 2 shfl_up per step)
//  - logits rows double-buffered in LDS via TDM, s_wait_tensorcnt sync
// ---------------------------------------------------------------------------
__global__ __launch_bounds__(32) void ctc_alpha(const float* __restrict__ yp,
                                                const int*   __restrict__ yt,
                                                const float* __restrict__ rowOff,
                                                float*       __restrict__ lossB) {
  __shared__ float rows[2][CH * Vv];   // 2 x 8KB tile buffers
  __shared__ float roff[Tt];           // 4KB per-t log-softmax offsets
  __shared__ int   labs[Ll];

  const int b    = blockIdx.x;
  const int lane = threadIdx.x;
  const float* base = yp + (size_t)b * Tt * Vv;

  // Kick off first two tile DMAs immediately.
  tdm_load_rows(base + 0 * CH * Vv, lds_off(&rows[0][0]));
  tdm_load_rows(base + 1 * CH * Vv, lds_off(&rows[1][0]));

  // Stage labels and per-row offsets into LDS (coalesced, one-time).
  #pragma unroll
  for (int i = lane; i < Ll; i += 32) labs[i] = yt[b * Ll + i];
  for (int i = lane; i < Tt; i += 32) roff[i] = rowOff[b * Tt + i];

  // Extended labels + skip/valid masks, register-resident per lane.
  int ext[NS];
  unsigned skipm = 0, validm = 0;
  #pragma unroll
  for (int j = 0; j < NS; j++) {
    int s = lane * NS + j;
    if (s < Ss) {
      validm |= 1u << j;
      if (s & 1) {
        int lab = labs[(s - 1) >> 1];
        ext[j] = lab;
        if (s >= 3 && lab != labs[(s - 3) >> 1]) skipm |= 1u << j;
      } else {
        ext[j] = BLANKC;
      }
    } else {
      ext[j] = 0; // safe dummy gather index
    }
  }

  float ap[NS];
  #pragma unroll
  for (int j = 0; j < NS; j++) ap[j] = NEGF;

  int t = 0;
  for (int c = 0; c < NCH; c++) {
    const int buf = c & 1;
    if (c == NCH - 1) { WAIT_TENSORCNT(0); }  // only this tile outstanding
    else              { WAIT_TENSORCNT(1); }  // oldest of two done

    for (int r = 0; r < CH; r++, t++) {
      const float ro = roff[t];
      float lp[NS];
      #pragma unroll
      for (int j = 0; j < NS; j++)
        lp[j] = rows[buf][r * Vv + ext[j]] - ro;   // LDS gather

      if (t == 0) {
        #pragma unroll
        for (int j = 0; j < NS; j++) {
          int s = lane * NS + j;
          ap[j] = (s < 2) ? lp[j] : NEGF;
        }
      } else {
        float p8 = __shfl_up(ap[NS - 1], 1, 32);   // state 9*lane-1
        float p7 = __shfl_up(ap[NS - 2], 1, 32);   // state 9*lane-2
        if (lane == 0) { p8 = NEGF; p7 = NEGF; }
        float an[NS];
        #pragma unroll
        for (int j = 0; j < NS; j++) {
          float a1  = ap[j];
          float a2  = (j == 0) ? p8 : ap[j - 1];
          float a3s = (j == 0) ? p7 : ((j == 1) ? p8 : ap[j - 2]);
          float a3  = ((skipm >> j) & 1) ? a3s : NEGF;
          float m = fmaxf(a1, fmaxf(a2, a3));
          float v;
          if (m <= NEGF) v = NEGF;
          else v = m + __logf(__expf(a1 - m) + __expf(a2 - m) + __expf(a3 - m));
          an[j] = ((validm >> j) & 1) ? (v + lp[j]) : NEGF;
        }
        #pragma unroll
        for (int j = 0; j < NS; j++) ap[j] = an[j];
      }
    }
    // This buffer fully consumed -> prefetch tile c+2 into it.
    if (c + 2 < NCH)
      tdm_load_rows(base + (size_t)(c + 2) * CH * Vv, lds_off(&rows[buf][0]));
  }

  // ll = logaddexp(alpha[S-1], alpha[S-2]); s=255 -> lane 28 j=3, s=256 -> lane 28 j=4
  float a255 = __shfl(ap[3], 28, 32);
  float a256 = __shfl(ap[4], 28, 32);
  float m  = fmaxf(a255, a256);
  float ll = (m <= NEGF) ? NEGF : m + __logf(__expf(a255 - m) + __expf(a256 - m));
  if (lane == 0) lossB[b] = -ll;
}

// ---------------------------------------------------------------------------
// Kernel 3: mean over B losses -> scalar
// ---------------------------------------------------------------------------
__global__ __launch_bounds__(32) void ctc_mean(const float* __restrict__ lossB,
                                               float* __restrict__ out) {
  const int lane = threadIdx.x;
  float s = 0.0f;
  #pragma unroll
  for (int i = lane; i < Bb; i += 32) s += lossB[i];
  #pragma unroll
  for (int d = 16; d >= 1; d >>= 1) s += __shfl_xor(s, d, 32);
  if (lane == 0) out[0] = s * (1.0f / Bb);
}

// ---------------------------------------------------------------------------
extern "C" void kernel_launch(void* const* d_in, const int* in_sizes, int n_in,
                              void* d_out, int out_size, void* d_ws, size_t ws_size,
                              hipStream_t stream) {
  const int*   y_true = (const int*)d_in[0];    // [B, L] int32
  const float* y_pred = (const float*)d_in[1];  // [B, T, V] f32
  float* rowOff = (float*)d_ws;                 // [B*T] f32
  float* lossB  = rowOff + (size_t)Bb * Tt;     // [B] f32

  ctc_row_lse<<<(Bb * Tt) / 8, 256, 0, stream>>>(y_pred, rowOff);
  ctc_alpha  <<<Bb,            32,  0, stream>>>(y_pred, y_true, rowOff, lossB);
  ctc_mean   <<<1,             32,  0, stream>>>(lossB, (float*)d_out);
}